// ScaledSum_50843822850395
// MI455X (gfx1250) — compile-verified
//
#include <hip/hip_runtime.h>
#include <stdint.h>

#define AS3 __attribute__((address_space(3)))
typedef float v4f __attribute__((ext_vector_type(4)));  // native clang vector: OK for NT builtins

// CDNA5 async DMA: global memory -> LDS, tracked by ASYNCcnt (no VGPR data).
__device__ __forceinline__ void async_copy_b128(AS3 float* lds_dst, const float* gsrc) {
  asm volatile("global_load_async_to_lds_b128 %0, %1, off"
               :: "v"(lds_dst), "v"(gsrc)
               : "memory");
}
__device__ __forceinline__ void wait_async_zero() {
  asm volatile("s_wait_asynccnt 0" ::: "memory");
}

// float offsets of P8,P16,P32,P64,P128,P256,P512 inside the workspace
__device__ constexpr size_t POFF[7] = {0, 1048576, 1310720, 1376256, 1392640, 1396736, 1397760};

// ---------------------------------------------------------------------------
// Pass 1: 8x8 box means. x:[8,512,512,32] f32 -> p8:[8,64,64,32] f32.
// Each 256-thread block (8 waves, wave32) owns an 8-row x 32-pixel x 32-ch
// tile. All 256 threads stage the tile into LDS with async b128 DMA (8 ops
// per thread), then 128 threads each reduce one 8x8 window for one channel.
// Pixel stride padded to 36 floats: keeps 16B alignment for the b128 LDS
// writes while breaking the 256-float (bank-aligned) stride between the four
// output pixels during the LDS reduce. x reads use default (RT) temporal
// policy on purpose: pass 2 re-reads x, and the tail of this stream stays
// resident in the 192MB L2.
// ---------------------------------------------------------------------------
__global__ void __launch_bounds__(256) pool8_kernel(const float* __restrict__ x,
                                                    float* __restrict__ p8) {
  constexpr int W = 512, C = 32;
  constexpr int PXS = 36;  // padded per-pixel stride in floats (144 B, 16B-aligned)
  __shared__ __align__(16) float tile[8 * 32 * PXS];

  const int blk = blockIdx.x;
  const int jg  = blk & 15;          // column group: 32 pixels = 4 output pixels
  const int i   = (blk >> 4) & 63;   // output row (covers 8 input rows)
  const int b   = blk >> 10;

  const int t  = threadIdx.x;
  const int c4 = t & 7;              // which float4 of the 32 channels
  const int px = t >> 3;             // 0..31 pixel within tile

  const float* base =
      x + (((size_t)b * 512 + (size_t)i * 8) * W + (size_t)jg * 32) * C;
  AS3 float* lt = (AS3 float*)tile;

#pragma unroll
  for (int row = 0; row < 8; ++row) {
    async_copy_b128(lt + (row * 32 + px) * PXS + c4 * 4,
                    base + (size_t)row * W * C + px * C + c4 * 4);
  }
  wait_async_zero();   // this wave's DMA landed in LDS
  __syncthreads();     // everyone's DMA landed

  if (t < 128) {
    const int c  = t & 31;
    const int jj = t >> 5;  // 0..3: output pixel within tile
    float s = 0.0f;
#pragma unroll
    for (int dy = 0; dy < 8; ++dy)
#pragma unroll
      for (int dx = 0; dx < 8; ++dx)
        s += tile[(dy * 32 + jj * 8 + dx) * PXS + c];
    p8[(((size_t)b * 64 + i) * 64 + (jg * 4 + jj)) * 32 + c] = s * (1.0f / 64.0f);
  }
}

// ---------------------------------------------------------------------------
// Hierarchy: build P16..P512 by repeated 2x2 means, one block per batch image,
// all levels in one launch (replaces 6 dependent micro-launches). Total
// traffic ~6.7 MiB, entirely L2-resident (P8 was just written). Levels are
// separated by a device fence + block barrier so each stage sees the
// previous stage's global writes.
// ---------------------------------------------------------------------------
__global__ void __launch_bounds__(256) pyramid_kernel(float* pyr) {
  const int b = blockIdx.x;   // 8 blocks
  const int t = threadIdx.x;  // 256 threads

  for (int s = 0; s < 6; ++s) {
    const int n = 32 >> s;                       // output dim: 32,16,8,4,2,1
    const v4f* in4  = (const v4f*)(pyr + POFF[s]);
    v4f*       out4 = (v4f*)(pyr + POFF[s + 1]);
    const int total = n * n * 8;                 // per-batch float4 outputs
    for (int k = t; k < total; k += 256) {
      const int c4 = k & 7;
      const int p  = k >> 3;
      const int wj = p % n;
      const int hi = p / n;
      const size_t row = (size_t)(2 * n) * 8;
      const size_t ib = ((size_t)(b * 2 * n + 2 * hi) * (2 * n) + 2 * wj) * 8 + c4;
      const v4f r = 0.25f * (in4[ib] + in4[ib + 8] + in4[ib + row] + in4[ib + row + 8]);
      out4[((size_t)(b * n + hi) * n + wj) * 8 + c4] = r;
    }
    __threadfence();   // flush stores to the device-coherent point (L2)
    __syncthreads();   // whole block before next level consumes them
  }
}

// ---------------------------------------------------------------------------
// Pass 2: out = (x + P8[h/8,w/8] + P16[h/16,w/16] + ... + P512) / 8.
// Traversal is REVERSED relative to pass 1's issue order so pass 2 consumes
// x in the order pass 1 finished writing L2 -> the hot tail (~192MB L2) of x
// is re-read from cache instead of HBM. x loads are last-use (non-temporal),
// out stores are non-temporal (never re-read) so the 256MiB output stream
// doesn't evict the x lines we still want. Pyramid (5.3 MiB) stays in L2;
// within a wave pyramid loads are 1 coalesced 128B line (or a broadcast).
// ---------------------------------------------------------------------------
__global__ void __launch_bounds__(256) final_kernel(const float* __restrict__ x,
                                                    const float* __restrict__ pyr,
                                                    float* __restrict__ out) {
  constexpr size_t TOTAL4 = (size_t)8 * 512 * 512 * 8;  // 16,777,216 float4s
  const size_t e = TOTAL4 - 1 - ((size_t)blockIdx.x * 256 + threadIdx.x);
  const int c4 = (int)(e & 7);
  const int w  = (int)((e >> 3) & 511);
  const int h  = (int)((e >> 12) & 511);
  const int b  = (int)(e >> 21);

  const v4f* x4 = (const v4f*)x;
  v4f acc = __builtin_nontemporal_load(&x4[e]);  // last use of this line

#pragma unroll
  for (int s = 0; s < 7; ++s) {
    const int shift = 3 + s;
    const int dim = 64 >> s;
    const v4f* p = (const v4f*)(pyr + POFF[s]);
    const size_t idx = (((size_t)b * dim + (h >> shift)) * dim + (w >> shift)) * 8 + c4;
    acc += p[idx];
  }
  const v4f r = acc * 0.125f;
  __builtin_nontemporal_store(r, &((v4f*)out)[e]);
}

extern "C" void kernel_launch(void* const* d_in, const int* in_sizes, int n_in,
                              void* d_out, int out_size, void* d_ws, size_t ws_size,
                              hipStream_t stream) {
  (void)in_sizes; (void)n_in; (void)out_size; (void)ws_size;
  const float* x = (const float*)d_in[0];
  // d_in[1] ("a") is softmax'ed over a size-1 axis -> all-ones weights; unused.
  float* out = (float*)d_out;
  float* pyr = (float*)d_ws;  // needs 1,398,016 floats (~5.34 MiB)

  // Pass 1: 8x8 box means into P8 (reads x once, HBM-bound).
  pool8_kernel<<<8 * 64 * 16, 256, 0, stream>>>(x, pyr);

  // Hierarchy: P8 -> P16 -> ... -> P512, one launch, one block per batch.
  pyramid_kernel<<<8, 256, 0, stream>>>(pyr);

  // Pass 2: fused sum of identity + 7 upsampled levels, scaled by 1/8,
  // traversed in reverse for L2 reuse of x.
  final_kernel<<<(8 * 512 * 512 * 8) / 256, 256, 0, stream>>>(x, pyr, out);
}